// GroupedIdxConv1D_47476568490423
// MI455X (gfx1250) — compile-verified
//
#include <hip/hip_runtime.h>

typedef __attribute__((ext_vector_type(2))) float v2f;
typedef __attribute__((ext_vector_type(8))) float v8f;

#define T_IN   2048
#define T_OUT  2042
#define CH     256
#define NRG    16
#define OUTD   64
#define KTAP   7
#define KPAD   8

__device__ __forceinline__ int imin(int a, int b) { return a < b ? a : b; }

// Channel-loop body: one K=7 tap-conv contribution as two K=4 f32 WMMAs.
// INTERIOR=true skips the edge clamp (all A indices provably in-bounds).
template <bool INTERIOR>
__device__ __forceinline__ void conv_step(const float* __restrict__ xr,
                                          const float* __restrict__ wl,
                                          int i0, int kb, v8f& acc) {
    v2f a0, a1;
    if (INTERIOR) {
        a0.x = xr[i0];     a0.y = xr[i0 + 1];
        a1.x = xr[i0 + 4]; a1.y = xr[i0 + 5];
    } else {
        // Clamp: out-of-range reads only feed rows t >= T_OUT (discarded).
        a0.x = xr[imin(i0,     T_IN - 1)];
        a0.y = xr[imin(i0 + 1, T_IN - 1)];
        a1.x = xr[imin(i0 + 4, T_IN - 1)];
        a1.y = xr[imin(i0 + 5, T_IN - 1)];
    }
    v2f b0, b1;
    b0.x = wl[kb];     b0.y = wl[kb + 1];     // taps 0..3
    b1.x = wl[4 + kb]; b1.y = wl[5 + kb];     // taps 4..6 (+zero pad)

    acc = __builtin_amdgcn_wmma_f32_16x16x4_f32(
              false, a0, false, b0, (short)0, acc, false, false);
    acc = __builtin_amdgcn_wmma_f32_16x16x4_f32(
              false, a1, false, b1, (short)0, acc, false, false);
}

// One wave computes a 16(t) x 16(d) f32 tile of out[b,:,:] via
// V_WMMA_F32_16X16X4_F32, accumulating over 256 channels x 7 taps.
__global__ __launch_bounds__(256)
void grouped_conv1d_wmma_f32(const float* __restrict__ x,
                             const long long* __restrict__ gidx,
                             const float* __restrict__ W,
                             const float* __restrict__ bias,
                             float* __restrict__ out) {
    // Stage all of W into LDS: [group][d][k] with k padded to 8, slot 7 = 0.
    __shared__ float Wlds[NRG * OUTD * KPAD];   // 32 KB
    const int tid = threadIdx.x;
    for (int i = tid; i < NRG * OUTD * KPAD; i += 256) {
        const int k = i & (KPAD - 1);
        const int d = (i >> 3) & (OUTD - 1);
        const int g = i >> 9;
        Wlds[i] = (k < KTAP) ? W[(g * OUTD + d) * KTAP + k] : 0.0f;
    }
    __syncthreads();

    const int lane = tid & 31;
    const int wave = tid >> 5;
    const int b    = blockIdx.y;

    // Force wave-uniform tile coordinates into SGPRs so the interior/edge
    // split is a scalar branch (no exec masking around the WMMA loop).
    const int d0 = __builtin_amdgcn_readfirstlane((wave & 3) * 16);
    const int t0 = __builtin_amdgcn_readfirstlane((blockIdx.x * 2 + (wave >> 2)) * 16);

    // f32 WMMA lane mapping (ISA 7.12.2):
    //   A 16x4 : lane holds row M = lane&15; VGPR0->K=2*(lane>=16), VGPR1->K+1
    //   B 4x16 : lane holds col N = lane&15; VGPR0->K=2*(lane>=16), VGPR1->K+1
    //   C 16x16: lane holds col N = lane&15; VGPR r -> M = r + 8*(lane>=16)
    const int m  = lane & 15;
    const int kb = (lane >> 4) * 2;
    const int n  = m;
    const int i0 = t0 + m + kb;

    const float*     xb  = x + (size_t)(b * CH) * T_IN;
    const long long* gb  = gidx + b * CH;
    const float*     wdn = &Wlds[(d0 + n) * KPAD];   // + g*OUTD*KPAD per channel

    v8f acc = {};
    int cnt0 = 0;

    if (t0 + 15 + 2 + 5 < T_IN) {            // uniform: all A reads in-bounds
        for (int c = 0; c < CH; ++c) {
            const int g = (int)gb[c];
            cnt0 += (g == 0) ? 1 : 0;
            conv_step<true>(xb + (size_t)c * T_IN, wdn + g * (OUTD * KPAD),
                            i0, kb, acc);
        }
    } else {                                  // edge tile (t0 == 2032 only)
        for (int c = 0; c < CH; ++c) {
            const int g = (int)gb[c];
            cnt0 += (g == 0) ? 1 : 0;
            conv_step<false>(xb + (size_t)c * T_IN, wdn + g * (OUTD * KPAD),
                             i0, kb, acc);
        }
    }

    // Epilogue: out[b,d,t] = acc + count0 * bias[d]
    const float bterm = bias[d0 + n] * (float)cnt0;
    float* orow = out + ((size_t)b * OUTD + (d0 + n)) * T_OUT;
    const int tb = t0 + ((lane >> 4) * 8);
#pragma unroll
    for (int r = 0; r < 8; ++r) {
        const int t = tb + r;
        if (t < T_OUT) orow[t] = acc[r] + bterm;
    }
}

extern "C" void kernel_launch(void* const* d_in, const int* in_sizes, int n_in,
                              void* d_out, int out_size, void* d_ws, size_t ws_size,
                              hipStream_t stream) {
    const float*     x    = (const float*)d_in[0];       // (16,256,2048) f32
    const long long* gidx = (const long long*)d_in[1];   // (16,256) int64
    const float*     W    = (const float*)d_in[2];       // (16,64,7) f32
    const float*     bias = (const float*)d_in[3];       // (64,) f32
    float*           out  = (float*)d_out;               // (16,64,2042) f32

    (void)in_sizes; (void)n_in; (void)out_size; (void)d_ws; (void)ws_size;

    dim3 grid(64, 16, 1);   // 64 t-blocks (2 t-tiles each), 16 batches
    dim3 block(256, 1, 1);  // 8 waves: 4 d-tiles x 2 t-tiles
    grouped_conv1d_wmma_f32<<<grid, block, 0, stream>>>(x, gidx, W, bias, out);
}